// HGCU_77292231458937
// MI455X (gfx1250) — compile-verified
//
#include <hip/hip_runtime.h>

// ---------------------------------------------------------------------------
// CARAFE upsampler for MI455X (gfx1250, wave32).
// GEMM stages use fp32 WMMA (V_WMMA_F32_16X16X4_F32) fed from LDS:
//   - activations staged once per block, read as conflict-free ds_load_b64
//   - weights repacked so A fragments are contiguous global b64 loads
//   - each wave computes two 16-pixel N-tiles sharing one A fragment
// Pointwise softmax / hamming-renorm and per-pixel CARAFE gathers are VALU.
// ---------------------------------------------------------------------------

typedef float v2f __attribute__((ext_vector_type(2)));
typedef float v4f __attribute__((ext_vector_type(4)));
typedef float v8f __attribute__((ext_vector_type(8)));

#define WMMA_F32(a, b, c) \
  __builtin_amdgcn_wmma_f32_16x16x4_f32(false, (a), false, (b), (short)0, (c), false, false)

// ----------------- conv1x1 as GEMM: D[oc][pix] = W[oc][c] * X[c][pix] -------
#define C1_CIN 256
#define C1_STR 260
__global__ void conv1x1_wmma_k(const float* __restrict__ x,
                               const float* __restrict__ w,
                               float* __restrict__ y,
                               int Cout, int HW) {
  __shared__ float sxx[32 * C1_STR];
  int tid  = threadIdx.x;
  int lane = tid & 31;
  int mt   = tid >> 5;
  int n0   = blockIdx.x * 32;
  int b    = n0 / HW;
  int hw0  = n0 - b * HW;

  const float* xb = x + ((size_t)b * C1_CIN) * HW + hw0;
  for (int g = tid; g < C1_CIN * 8; g += 128) {
    int c  = g >> 3;
    int p4 = (g & 7) << 2;
    v4f v = *(const v4f*)(xb + (size_t)c * HW + p4);
#pragma unroll
    for (int j = 0; j < 4; ++j) sxx[(p4 + j) * C1_STR + c] = v[j];
  }
  __syncthreads();

  int lm = lane & 15;
  int kh = (lane >> 4) << 1;
  int m0 = mt << 4;
  const float* wr = w + (size_t)(m0 + lm) * C1_CIN + kh;
  const float* s0 = &sxx[lm * C1_STR + kh];
  const float* s1 = &sxx[(lm + 16) * C1_STR + kh];

  v8f acc0 = {}, acc1 = {};
#pragma unroll 4
  for (int k0 = 0; k0 < C1_CIN; k0 += 4) {
    v2f a  = *(const v2f*)(wr + k0);
    v2f b0 = *(const v2f*)(s0 + k0);
    v2f b1 = *(const v2f*)(s1 + k0);
    acc0 = WMMA_F32(a, b0, acc0);
    acc1 = WMMA_F32(a, b1, acc1);
  }

  float* yb = y + ((size_t)b * Cout) * HW + hw0 + lm;
  int mhi = (lane >> 4) << 3;
#pragma unroll
  for (int r = 0; r < 8; ++r) {
    yb[(size_t)(m0 + mhi + r) * HW]      = acc0[r];
    yb[(size_t)(m0 + mhi + r) * HW + 16] = acc1[r];
  }
}

// ---- weight repack for conv3x3: w[oc][c][t] -> wt[oc][t][c] ----------------
__global__ void repack_w_k(const float* __restrict__ w, float* __restrict__ wt,
                           int total) {
  int i = blockIdx.x * blockDim.x + threadIdx.x;
  if (i >= total) return;
  int c = i & 63;
  int rest = i >> 6;
  int t  = rest % 9;
  int oc = rest / 9;
  wt[i] = w[((size_t)oc * 64 + c) * 9 + t];
}

// -------- conv3x3 (zero pad) as LDS-fed WMMA GEMM ---------------------------
#define C3_COLS 34
#define C3_STR  68
__global__ void conv3x3_wmma_k(const float* __restrict__ x,
                               const float* __restrict__ wt,
                               const float* __restrict__ bias,
                               float* __restrict__ y,
                               int Cout, int H, int W) {
  __shared__ float sx[3 * C3_COLS * C3_STR];
  const int Cin = 64;
  int tid  = threadIdx.x;
  int lane = tid & 31;
  int mt   = tid >> 5;
  int tpr  = W >> 5;
  int bid  = blockIdx.x;
  int wtl  = bid % tpr;
  int hh   = (bid / tpr) % H;
  int b    = bid / (tpr * H);
  int w0   = wtl << 5;
  int HWc  = H * W;

  const float* xb = x + (size_t)b * Cin * HWc;
  for (int idx = tid; idx < 3 * C3_COLS * Cin; idx += 224) {
    int col  = idx % C3_COLS;
    int rest = idx / C3_COLS;
    int c    = rest & 63;
    int ky   = rest >> 6;
    int sy   = hh + ky - 1;
    int gx   = w0 + col - 1;
    float v  = 0.0f;
    if (sy >= 0 && sy < H && gx >= 0 && gx < W)
      v = xb[(size_t)c * HWc + (size_t)sy * W + gx];
    sx[(ky * C3_COLS + col) * C3_STR + c] = v;
  }
  __syncthreads();

  int lm = lane & 15;
  int kh = (lane >> 4) << 1;
  int m0 = mt << 4;
  int oc = m0 + lm;
  if (oc >= Cout) oc = Cout - 1;
  const float* wr = wt + (size_t)oc * 9 * 64 + kh;

  v8f acc0 = {}, acc1 = {};
  for (int t = 0; t < 9; ++t) {
    int ky = t / 3, kx = t - 3 * ky;
    const float* wp = wr + t * 64;
    const float* s0 = &sx[(ky * C3_COLS + lm + kx) * C3_STR + kh];
#pragma unroll 4
    for (int c0 = 0; c0 < 64; c0 += 4) {
      v2f a  = *(const v2f*)(wp + c0);
      v2f b0 = *(const v2f*)(s0 + c0);
      v2f b1 = *(const v2f*)(s0 + 16 * C3_STR + c0);
      acc0 = WMMA_F32(a, b0, acc0);
      acc1 = WMMA_F32(a, b1, acc1);
    }
  }

  int mhi = (lane >> 4) << 3;
  float* yb = y + (size_t)b * Cout * HWc + (size_t)hh * W + w0 + lm;
#pragma unroll
  for (int r = 0; r < 8; ++r) {
    int m = m0 + mhi + r;
    if (m < Cout) {
      float bv = bias[m];
      yb[(size_t)m * HWc]      = acc0[r] + bv;
      yb[(size_t)m * HWc + 16] = acc1[r] + bv;
    }
  }
}

// ---- normalize_kernel phase A: softmax over 25 channels (stride HW) --------
__global__ void softmax25_k(const float* __restrict__ src,
                            float* __restrict__ dst, int HW, int total) {
  int idx = blockIdx.x * blockDim.x + threadIdx.x;
  if (idx >= total) return;
  int q  = idx % HW;
  int bg = idx / HW;
  const float* p = src + (size_t)bg * 25 * HW + q;
  float v[25];
  float mx = -3.0e38f;
#pragma unroll
  for (int c = 0; c < 25; ++c) { v[c] = p[(size_t)c * HW]; mx = fmaxf(mx, v[c]); }
  float s = 0.f;
#pragma unroll
  for (int c = 0; c < 25; ++c) { v[c] = __expf(v[c] - mx); s += v[c]; }
  float inv = 1.0f / s;
  float* o = dst + (size_t)bg * 25 * HW + q;
#pragma unroll
  for (int c = 0; c < 25; ++c) o[(size_t)c * HW] = v[c] * inv;
}

// ---- normalize_kernel phase B: faithful raw-reinterpret hamming renorm -----
__global__ void renorm25_k(const float* __restrict__ s,
                           const float* __restrict__ ham,
                           float* __restrict__ out, int HW, int total) {
  int idx = blockIdx.x * blockDim.x + threadIdx.x;
  if (idx >= total) return;
  int q  = idx % HW;
  int bg = idx / HW;
  const float* p = s + (size_t)bg * 25 * HW + (size_t)q * 25;
  float v[25];
  float sum = 0.f;
#pragma unroll
  for (int t = 0; t < 25; ++t) { v[t] = p[t] * ham[t]; sum += v[t]; }
  float inv = 1.0f / (sum + 1e-8f);
  float* o = out + (size_t)bg * 25 * HW + q;
#pragma unroll
  for (int t = 0; t < 25; ++t) o[(size_t)t * HW] = v[t] * inv;
}

// ---- CARAFE: out[b,c,oh,ow] = (add?) + sum_k x[b,c,refl(nb(k))] * mavg -----
__global__ void carafe_k(const float* __restrict__ x, int Cx,
                         const float* __restrict__ mask,
                         const float* __restrict__ addsrc,
                         float* __restrict__ out,
                         int lH, int lW, int OH, int OW) {
  __shared__ float mavg[25 * 64];
  int tid = threadIdx.x;
  int bpr = OW >> 6;
  int bid = blockIdx.x;
  int owt = bid % bpr;
  int oh  = (bid / bpr) % OH;
  int b   = bid / (bpr * OH);
  int ow0 = owt << 6;
  int OHW = OH * OW;
  int lHW = lH * lW;

  const float* mb = mask + (size_t)b * 100 * OHW + (size_t)oh * OW + ow0;
  for (int t = tid; t < 25 * 64; t += blockDim.x) {
    int k = t >> 6, owl = t & 63;
    float s = 0.f;
#pragma unroll
    for (int g = 0; g < 4; ++g) s += mb[(size_t)(g * 25 + k) * OHW + owl];
    mavg[t] = 0.25f * s;
  }
  __syncthreads();

  int owl = tid & 63;
  int cst = tid >> 6;
  int ow  = ow0 + owl;
  int lh  = oh >> 1, lw = ow >> 1;

  int syy[5], sxx2[5];
#pragma unroll
  for (int i = 0; i < 5; ++i) {
    int t2 = lh + i - 2; if (t2 < 0) t2 = -t2;
    if (t2 > lH - 1) t2 = 2 * (lH - 1) - t2; syy[i] = t2;
  }
#pragma unroll
  for (int j = 0; j < 5; ++j) {
    int t2 = lw + j - 2; if (t2 < 0) t2 = -t2;
    if (t2 > lW - 1) t2 = 2 * (lW - 1) - t2; sxx2[j] = t2;
  }

  const float* xb = x + (size_t)b * Cx * lHW;
  const float* ab = addsrc ? addsrc + (size_t)b * Cx * OHW + (size_t)oh * OW + ow
                           : nullptr;
  float* ob = out + (size_t)b * Cx * OHW + (size_t)oh * OW + ow;

  for (int c = cst; c < Cx; c += 4) {
    const float* xc = xb + (size_t)c * lHW;
    if (c + 4 < Cx)
      __builtin_prefetch(xc + (size_t)4 * lHW + (size_t)syy[0] * lW, 0, 1);
    float acc = ab ? ab[(size_t)c * OHW] : 0.0f;
#pragma unroll
    for (int i = 0; i < 5; ++i) {
      const float* xr = xc + (size_t)syy[i] * lW;
#pragma unroll
      for (int j = 0; j < 5; ++j)
        acc += xr[sxx2[j]] * mavg[(i * 5 + j) * 64 + owl];
    }
    ob[(size_t)c * OHW] = acc;
  }
}

extern "C" void kernel_launch(void* const* d_in, const int* in_sizes, int n_in,
                              void* d_out, int out_size, void* d_ws, size_t ws_size,
                              hipStream_t stream) {
  (void)in_sizes; (void)n_in; (void)out_size; (void)ws_size;
  const float* hr    = (const float*)d_in[0];
  const float* lr    = (const float*)d_in[1];
  const float* w_hr  = (const float*)d_in[2];
  const float* w_lr  = (const float*)d_in[3];
  const float* wk_hr = (const float*)d_in[4];
  const float* bk_hr = (const float*)d_in[5];
  const float* wk_lr = (const float*)d_in[6];
  const float* bk_lr = (const float*)d_in[7];
  const float* ham   = (const float*)d_in[8];
  float* out = (float*)d_out;
  float* ws  = (float*)d_ws;

  const int B = 2, Chr = 256, Cc = 64, KC = 100;
  const int H = 128, W = 128, hL = 64, wL = 64;
  const int HWh = H * W, HWl = hL * wL;
  const int WPK = KC * 9 * 64;

  float* c_hr  = ws;
  float* c_lr  = c_hr  + (size_t)B * Cc * HWh;
  float* k_hr  = c_lr  + (size_t)B * Cc * HWl;
  float* k_lr  = k_hr  + (size_t)B * KC * HWh;
  float* t0    = k_lr  + (size_t)B * KC * HWl;
  float* t1    = t0    + (size_t)B * KC * HWh;
  float* wt_hr = t1    + (size_t)B * KC * HWh;
  float* wt_lr = wt_hr + WPK;

  repack_w_k<<<(WPK + 255) / 256, 256, 0, stream>>>(wk_hr, wt_hr, WPK);
  repack_w_k<<<(WPK + 255) / 256, 256, 0, stream>>>(wk_lr, wt_lr, WPK);

  conv1x1_wmma_k<<<(B * HWh) / 32, 128, 0, stream>>>(hr, w_hr, c_hr, Cc, HWh);
  conv1x1_wmma_k<<<(B * HWl) / 32, 128, 0, stream>>>(lr, w_lr, c_lr, Cc, HWl);

  conv3x3_wmma_k<<<B * H * (W / 32), 224, 0, stream>>>(c_hr, wt_hr, bk_hr, k_hr, KC, H, W);
  conv3x3_wmma_k<<<B * hL * (wL / 32), 224, 0, stream>>>(c_lr, wt_lr, bk_lr, k_lr, KC, hL, wL);

  int tot = B * 4 * HWh;
  int nb  = (tot + 255) / 256;

  softmax25_k<<<nb, 256, 0, stream>>>(k_hr, t0, HWh, tot);
  renorm25_k<<<nb, 256, 0, stream>>>(t0, ham, t1, HWh, tot);

  carafe_k<<<B * H * (W / 64), 256, 0, stream>>>(k_lr, KC, t1, k_hr, t0, hL, wL, H, W);

  softmax25_k<<<nb, 256, 0, stream>>>(t0, t1, HWh, tot);
  renorm25_k<<<nb, 256, 0, stream>>>(t1, ham, k_hr, HWh, tot);

  carafe_k<<<B * H * (W / 64), 256, 0, stream>>>(lr, Chr, k_hr, nullptr, out, hL, wL, H, W);
}